// ChannelAttention_32109175505505
// MI455X (gfx1250) — compile-verified
//
#include <hip/hip_runtime.h>
#include <math.h>

// Problem constants (fixed by the reference)
#define B_     2048
#define C_     16
#define S_     2000
#define E_     20       // S/INTER
#define INTER_ 100
#define EPS_   1e-5f
#define NTILE  125      // position-tiles of 16 per batch row
#define NWAVES 8        // waves per block (256 threads)

typedef __attribute__((ext_vector_type(2))) float v2f;
typedef __attribute__((ext_vector_type(8))) float v8f;

__device__ __forceinline__ v2f make2(float a, float b) { v2f r; r.x = a; r.y = b; return r; }

// One K=4 slice of a 16x16x16 f32 matmul on the CDNA5 matrix pipe.
// A = M x K (row within lane, M = lane%16), B = K x N (N = lane%16),
// D[M][N]: N = lane%16, M = vgpr + 8*(lane>=16).
__device__ __forceinline__ v8f wmma4(v2f a, v2f b, v8f c) {
  return __builtin_amdgcn_wmma_f32_16x16x4_f32(
      /*neg_a=*/false, a, /*neg_b=*/false, b,
      /*c_mod=*/(short)0, c, /*reuse_a=*/false, /*reuse_b=*/false);
}

// Combine the two wave32 halves (lane l <-> l^16 hold the same D column).
__device__ __forceinline__ float xhalf_sum(float v) { return v + __shfl_xor(v, 16, 32); }
__device__ __forceinline__ float xhalf_max(float v) { return fmaxf(v, __shfl_xor(v, 16, 32)); }

// A-operand fragment of W' = W - colmean(W): folding the LayerNorm mean into
// the weights makes mean_j(x @ W'^T + b') == 0 analytically, so LN needs only
// a sum of squares. Lane's A row M = lane%16 (out-channel); K cols {c0, c0+1}.
__device__ __forceinline__ v2f wfrag_centered(const float* __restrict__ W,
                                              int l16, int c0) {
  float s0 = 0.f, s1 = 0.f;
#pragma unroll
  for (int j = 0; j < 16; ++j) { s0 += W[j * C_ + c0]; s1 += W[j * C_ + c0 + 1]; }
  return make2(W[l16 * C_ + c0] - s0 * (1.f / 16.f),
               W[l16 * C_ + c0 + 1] - s1 * (1.f / 16.f));
}

__device__ __forceinline__ float vmean16(const float* __restrict__ v) {
  float s = 0.f;
#pragma unroll
  for (int j = 0; j < 16; ++j) s += v[j];
  return s * (1.f / 16.f);
}

// ---------------------------------------------------------------------------
// Fully fused channel attention: one block (256 thr / 8 waves) per batch row.
// x[b] (128 KB) is read from HBM exactly once; Phase A stages it into LDS
// (fits easily in CDNA5's 320 KB WGP LDS) so Phase C never touches HBM for x.
// Total traffic = 262 MB read + 262 MB write = the roofline floor.
//
// Phase A: q/k = mean-folded LN(x^T W'{q,k}^T + b') via chained f32 WMMAs;
//          AvgPool(100) sums accumulate in LDS (ds_add_f32).
// Phase B: wave 0: atten^T = ck cq^T (5 WMMAs; AvgPool's 1/100^2 and
//          1/sqrt(E) folded into one logit constant), in-lane softmax +
//          rowsum over m -> per-channel scale.
// Phase C: out = LN((x*scale)^T Wp'^T + bp'); the channel scale is folded
//          into the centered Wp columns at setup, x comes from LDS, stores
//          are direct coalesced 64B row segments.
// ---------------------------------------------------------------------------
__global__ __launch_bounds__(256) void k_fused(
    const float* __restrict__ x,
    const float* __restrict__ Wq, const float* __restrict__ bq,
    const float* __restrict__ gq, const float* __restrict__ hq,
    const float* __restrict__ Wk, const float* __restrict__ bk,
    const float* __restrict__ gk, const float* __restrict__ hk,
    const float* __restrict__ Wp, const float* __restrict__ bp,
    const float* __restrict__ gp, const float* __restrict__ hp,
    float* __restrict__ out) {
  const int b    = blockIdx.x;
  const int tid  = threadIdx.x;
  const int lane = tid & 31;
  const int wave = tid >> 5;
  const int l16  = lane & 15;   // A row (out-channel) / B,D column (position)
  const int h    = lane >> 4;   // wave32 half: D rows (channels) 0-7 vs 8-15

  __shared__ float xs[C_][S_];       // staged x[b]: 128 KB of 320 KB WGP LDS
  __shared__ float sq[C_ * E_];      // pooled q sums [ch][e]
  __shared__ float sk[C_ * E_];      // pooled k sums [ch][e]
  __shared__ float scale_s[C_];      // rowsum(softmax(atten)) per channel
  for (int i = tid; i < C_ * E_; i += 256) { sq[i] = 0.f; sk[i] = 0.f; }
  __syncthreads();

  // ---- Phase A setup: centered weights + per-lane LN parameters -----------
  v2f wqf[4], wkf[4];
#pragma unroll
  for (int i = 0; i < 4; ++i) {
    const int c0 = 4 * i + 2 * h;
    wqf[i] = wfrag_centered(Wq, l16, c0);
    wkf[i] = wfrag_centered(Wk, l16, c0);
  }
  const float bmq = vmean16(bq), bmk = vmean16(bk);
  float bq8[8], gq8[8], hq8[8], bk8[8], gk8[8], hk8[8];
#pragma unroll
  for (int r = 0; r < 8; ++r) {
    const int ch = r + 8 * h;        // this lane's D-row channel for vgpr r
    bq8[r] = bq[ch] - bmq; gq8[r] = gq[ch]; hq8[r] = hq[ch];
    bk8[r] = bk[ch] - bmk; gk8[r] = gk[ch]; hk8[r] = hk[ch];
  }

  const float* xb = x + (size_t)b * C_ * S_;

  // ---- Phase A: q/k WMMA chains + LN + pooling; stage x into LDS ----------
  for (int t = wave; t < NTILE; t += NWAVES) {
    const int p = t * 16 + l16;
    const int e = p / INTER_;
    if (t + NWAVES < NTILE)  // speculative prefetch (global_prefetch_b8)
      __builtin_prefetch(&xb[(size_t)(2 * h) * S_ + (t + NWAVES) * 16 + l16], 0, 1);

    // B fragments: B[k=channel][n=position], per-lane K rows {c0, c0+1}.
    v2f xf[4];
#pragma unroll
    for (int i = 0; i < 4; ++i) {
      const int c0 = 4 * i + 2 * h;
      xf[i] = make2(xb[(size_t)c0 * S_ + p], xb[(size_t)(c0 + 1) * S_ + p]);
      xs[c0][p]     = xf[i].x;       // stage for Phase C (bank-conflict-free)
      xs[c0 + 1][p] = xf[i].y;
    }
    v8f dq = {}; v8f dk = {};
#pragma unroll
    for (int i = 0; i < 4; ++i) dq = wmma4(wqf[i], xf[i], dq);
#pragma unroll
    for (int i = 0; i < 4; ++i) dk = wmma4(wkf[i], xf[i], dk);

    {  // q: zero-mean LN (mean folded into W'), then pooled accumulation
      float d[8], ssq = 0.f;
#pragma unroll
      for (int r = 0; r < 8; ++r) { d[r] = dq[r] + bq8[r]; ssq = fmaf(d[r], d[r], ssq); }
      const float inv = __builtin_amdgcn_rsqf(xhalf_sum(ssq) * (1.f / 16.f) + EPS_);
#pragma unroll
      for (int r = 0; r < 8; ++r)
        atomicAdd(&sq[(r + 8 * h) * E_ + e], fmaf(d[r] * inv, gq8[r], hq8[r]));
    }
    {  // k
      float d[8], ssq = 0.f;
#pragma unroll
      for (int r = 0; r < 8; ++r) { d[r] = dk[r] + bk8[r]; ssq = fmaf(d[r], d[r], ssq); }
      const float inv = __builtin_amdgcn_rsqf(xhalf_sum(ssq) * (1.f / 16.f) + EPS_);
#pragma unroll
      for (int r = 0; r < 8; ++r)
        atomicAdd(&sk[(r + 8 * h) * E_ + e], fmaf(d[r] * inv, gk8[r], hk8[r]));
    }
  }
  __syncthreads();

  // ---- Phase B: attention logits + softmax rowsum (wave 0 only) -----------
  if (wave == 0) {
    v8f acc = {};
#pragma unroll
    for (int i = 0; i < 5; ++i) {
      const int e0 = 4 * i + 2 * h;
      v2f a  = make2(sk[l16 * E_ + e0], sk[l16 * E_ + e0 + 1]);  // A[m=l16][e]
      v2f bm = make2(sq[l16 * E_ + e0], sq[l16 * E_ + e0 + 1]);  // B[e][c=l16]
      acc = wmma4(a, bm, acc);
    }
    // D[m=r+8h][c=l16]; logit scale folds AvgPool's (1/100)^2 and 1/sqrt(20):
    const float logit_scale = 2.23606797e-5f;
    float v[8], mx = -3.4e38f;
#pragma unroll
    for (int r = 0; r < 8; ++r) { v[r] = acc[r] * logit_scale; mx = fmaxf(mx, v[r]); }
    mx = xhalf_max(mx);
    float ex[8], se = 0.f;
#pragma unroll
    for (int r = 0; r < 8; ++r) { ex[r] = __expf(v[r] - mx); se += ex[r]; }
    se = xhalf_sum(se);
    float rs = 0.f;
#pragma unroll
    for (int r = 0; r < 8; ++r) rs += ex[r] / se;  // softmax elements
    rs = xhalf_sum(rs);                            // faithful rowsum (~1.0)
    if (h == 0) scale_s[l16] = rs;
  }
  __syncthreads();

  // ---- Phase C setup: fold channel scale into centered Wp columns ---------
  v2f wpf[4];
#pragma unroll
  for (int i = 0; i < 4; ++i) {
    const int c0 = 4 * i + 2 * h;
    v2f w  = wfrag_centered(Wp, l16, c0);
    wpf[i] = make2(w.x * scale_s[c0], w.y * scale_s[c0 + 1]);
  }
  const float bmp = vmean16(bp);
  float bp8[8], gp8[8], hp8[8];
#pragma unroll
  for (int r = 0; r < 8; ++r) {
    const int ch = r + 8 * h;
    bp8[r] = bp[ch] - bmp; gp8[r] = gp[ch]; hp8[r] = hp[ch];
  }

  float* ob = out + (size_t)b * C_ * S_;

  // ---- Phase C: final projection + LN, x fragments from LDS ---------------
  for (int t = wave; t < NTILE; t += NWAVES) {
    const int p = t * 16 + l16;
    v8f acc = {};
#pragma unroll
    for (int i = 0; i < 4; ++i) {
      const int c0 = 4 * i + 2 * h;
      v2f xf = make2(xs[c0][p], xs[c0 + 1][p]);
      acc = wmma4(wpf[i], xf, acc);
    }
    float d[8], ssq = 0.f;
#pragma unroll
    for (int r = 0; r < 8; ++r) { d[r] = acc[r] + bp8[r]; ssq = fmaf(d[r], d[r], ssq); }
    const float inv = __builtin_amdgcn_rsqf(xhalf_sum(ssq) * (1.f / 16.f) + EPS_);
    // Row channel = r + 8h, positions t*16..t*16+15 across each half's
    // lanes -> two contiguous 64B segments per store instruction.
#pragma unroll
    for (int r = 0; r < 8; ++r)
      ob[(size_t)(r + 8 * h) * S_ + p] = fmaf(d[r] * inv, gp8[r], hp8[r]);
  }
}

extern "C" void kernel_launch(void* const* d_in, const int* in_sizes, int n_in,
                              void* d_out, int out_size, void* d_ws, size_t ws_size,
                              hipStream_t stream) {
  const float* x  = (const float*)d_in[0];
  const float* Wq = (const float*)d_in[1];
  const float* bq = (const float*)d_in[2];
  const float* gq = (const float*)d_in[3];
  const float* hq = (const float*)d_in[4];
  const float* Wk = (const float*)d_in[5];
  const float* bk = (const float*)d_in[6];
  const float* gk = (const float*)d_in[7];
  const float* hk = (const float*)d_in[8];
  const float* Wp = (const float*)d_in[9];
  const float* bp = (const float*)d_in[10];
  const float* gp = (const float*)d_in[11];
  const float* hp = (const float*)d_in[12];
  float* out = (float*)d_out;
  (void)d_ws; (void)ws_size; (void)in_sizes; (void)n_in; (void)out_size;

  k_fused<<<B_, 256, 0, stream>>>(x, Wq, bq, gq, hq, Wk, bk, gk, hk,
                                  Wp, bp, gp, hp, out);
}